// PairForce_50757923504449
// MI455X (gfx1250) — compile-verified
//
#include <hip/hip_runtime.h>

// Lennard-Jones pair energy + autograd forces (fused fwd+bwd) for MI455X.
// Memory/atomic-bound gather-scatter: no matrix structure -> WMMA inapplicable.
// Strategy: NT-hinted b128 index streaming, batched L2-cached pos gathers
// (all 8 gathers in flight before the FLOP chain), no-return relaxed
// agent-scope f32 atomics for the scatter, wave32 tree reduction for energy.
//
// NOTE: lj_edges_kernel is defined FIRST so the disasm snippet shows the hot
// kernel (verify global_atomic_add_f32, clustered gather loads).

typedef int int4v __attribute__((ext_vector_type(4)));

__device__ __forceinline__ void lj_atomic_add(float* p, float v) {
    // Relaxed, agent scope, result unused -> global_atomic_add_f32 (no-return).
    __hip_atomic_fetch_add(p, v, __ATOMIC_RELAXED, __HIP_MEMORY_SCOPE_AGENT);
}

__device__ __forceinline__ void lj_edge_scalar(const float* __restrict__ pos,
                                               float sig2, float eps,
                                               int s, int d,
                                               float* __restrict__ F,
                                               float& eacc) {
    const float* ps = pos + 3 * s;
    const float* pd = pos + 3 * d;
    float dx = ps[0] - pd[0];
    float dy = ps[1] - pd[1];
    float dz = ps[2] - pd[2];
    float r2   = fmaf(dx, dx, fmaf(dy, dy, dz * dz));
    float ir2  = 1.0f / r2;
    float s2   = sig2 * ir2;
    float sr6  = s2 * s2 * s2;
    float sr12 = sr6 * sr6;
    eacc += 4.0f * eps * (sr12 - sr6);
    float h  = 12.0f * eps * (2.0f * sr12 - sr6) * ir2;   // = -c/2
    float fx = h * dx, fy = h * dy, fz = h * dz;
    lj_atomic_add(&F[3 * s + 0],  fx);
    lj_atomic_add(&F[3 * s + 1],  fy);
    lj_atomic_add(&F[3 * s + 2],  fz);
    lj_atomic_add(&F[3 * d + 0], -fx);
    lj_atomic_add(&F[3 * d + 1], -fy);
    lj_atomic_add(&F[3 * d + 2], -fz);
}

__global__ void __launch_bounds__(256)
lj_edges_kernel(const float* __restrict__ pos,
                const float* __restrict__ sig_p,
                const float* __restrict__ eps_p,
                const int*   __restrict__ ei,
                float* __restrict__ out,   // [0]=energy, [1..3N]=forces
                int E, int vec4) {
    const float sigma = sig_p[0];
    const float eps   = eps_p[0];
    const float sig2  = sigma * sigma;
    float* __restrict__ F = out + 1;
    const int* __restrict__ src = ei;
    const int* __restrict__ dst = ei + E;

    const int tid = blockIdx.x * blockDim.x + threadIdx.x;
    float eacc = 0.0f;

    if (vec4) {
        // Main path (E % 4 == 0): one int4 of src + one int4 of dst per
        // thread, streamed non-temporally (index stream is touched once;
        // keep pos/force hot in L2 instead).
        const int E4 = E >> 2;
        if (tid < E4) {
            int4v s4 = __builtin_nontemporal_load((const int4v*)src + tid);
            int4v d4 = __builtin_nontemporal_load((const int4v*)dst + tid);

            // ---- Phase 1: issue ALL 8 atom gathers before any compute ----
            float psx[4], psy[4], psz[4], pdx[4], pdy[4], pdz[4];
#pragma unroll
            for (int k = 0; k < 4; ++k) {
                const float* ps = pos + 3 * s4[k];
                const float* pd = pos + 3 * d4[k];
                psx[k] = ps[0]; psy[k] = ps[1]; psz[k] = ps[2];
                pdx[k] = pd[0]; pdy[k] = pd[1]; pdz[k] = pd[2];
            }

            // ---- Phase 2: FLOP chain ----
            float fx[4], fy[4], fz[4];
#pragma unroll
            for (int k = 0; k < 4; ++k) {
                float dx = psx[k] - pdx[k];
                float dy = psy[k] - pdy[k];
                float dz = psz[k] - pdz[k];
                float r2   = fmaf(dx, dx, fmaf(dy, dy, dz * dz));
                float ir2  = 1.0f / r2;
                float s2   = sig2 * ir2;
                float sr6  = s2 * s2 * s2;
                float sr12 = sr6 * sr6;
                eacc += 4.0f * eps * (sr12 - sr6);
                float h = 12.0f * eps * (2.0f * sr12 - sr6) * ir2;  // -c/2
                fx[k] = h * dx; fy[k] = h * dy; fz[k] = h * dz;
            }

            // ---- Phase 3: scatter burst (24 no-return f32 atomics) ----
#pragma unroll
            for (int k = 0; k < 4; ++k) {
                lj_atomic_add(&F[3 * s4[k] + 0],  fx[k]);
                lj_atomic_add(&F[3 * s4[k] + 1],  fy[k]);
                lj_atomic_add(&F[3 * s4[k] + 2],  fz[k]);
                lj_atomic_add(&F[3 * d4[k] + 0], -fx[k]);
                lj_atomic_add(&F[3 * d4[k] + 1], -fy[k]);
                lj_atomic_add(&F[3 * d4[k] + 2], -fz[k]);
            }
        }
    } else {
        // Fallback: scalar grid-stride (only when E % 4 != 0).
        const int stride = gridDim.x * blockDim.x;
        for (int e = tid; e < E; e += stride)
            lj_edge_scalar(pos, sig2, eps, src[e], dst[e], F, eacc);
    }

    // ---- energy reduction: wave32 shuffle tree, then LDS across 8 waves ----
    for (int off = 16; off > 0; off >>= 1)
        eacc += __shfl_down(eacc, off, 32);

    __shared__ float wsum[8];               // 256 threads = 8 wave32
    const int lane = threadIdx.x & 31;
    const int wv   = threadIdx.x >> 5;
    if (lane == 0) wsum[wv] = eacc;
    __syncthreads();
    if (wv == 0) {
        float v = (lane < 8) ? wsum[lane] : 0.0f;
        for (int off = 4; off > 0; off >>= 1)
            v += __shfl_down(v, off, 32);
        if (lane == 0)
            lj_atomic_add(&out[0], v);
    }
}

__global__ void lj_zero_kernel(float* __restrict__ out, int n) {
    int i = blockIdx.x * blockDim.x + threadIdx.x;
    if (i < n) out[i] = 0.0f;
}

extern "C" void kernel_launch(void* const* d_in, const int* in_sizes, int n_in,
                              void* d_out, int out_size, void* d_ws, size_t ws_size,
                              hipStream_t stream) {
    (void)n_in; (void)d_ws; (void)ws_size;
    const float* pos = (const float*)d_in[0];   // [N,3] f32
    const float* sig = (const float*)d_in[1];   // scalar f32
    const float* eps = (const float*)d_in[2];   // scalar f32
    const int*   ei  = (const int*)d_in[3];     // [2,E] i32
    float* out = (float*)d_out;                 // [0]=energy, [1..3N]=forces

    const int E = in_sizes[3] / 2;

    // 1) zero energy + force accumulators (harness poisons d_out).
    {
        int blk = 256;
        int grd = (out_size + blk - 1) / blk;
        if (grd < 1) grd = 1;
        lj_zero_kernel<<<grd, blk, 0, stream>>>(out, out_size);
    }

    // 2) fused energy + force pass over all edges.
    {
        const int vec4 = ((E & 3) == 0) ? 1 : 0;
        int work = vec4 ? (E >> 2) : E;
        if (work < 1) work = 1;
        int blk = 256;
        int grd = (work + blk - 1) / blk;
        if (!vec4 && grd > 16384) grd = 16384;  // grid-stride fallback cap
        lj_edges_kernel<<<grd, blk, 0, stream>>>(pos, sig, eps, ei, out, E, vec4);
    }
}